// LRA_GNN_63024350102325
// MI455X (gfx1250) — compile-verified
//
#include <hip/hip_runtime.h>
#include <hip/hip_bf16.h>
#include <stdint.h>

#define N_NODES 100000
#define N_EDGES 1600000
#define HIDC    64
#define HEADS   8
#define HEAD_D  8
#define N_LAYERS 12
#define OUTC    10
#define RW_STEPS 5
#define NEG_SLOPE 0.2f

typedef __attribute__((ext_vector_type(2))) float        v2f;
typedef __attribute__((ext_vector_type(8))) float        v8f;
typedef __attribute__((ext_vector_type(4))) unsigned int v4u;
typedef __attribute__((ext_vector_type(8))) int          v8i;
typedef __attribute__((ext_vector_type(4))) int          v4i;

__device__ __forceinline__ float leaky(float v) { return v > 0.f ? v : NEG_SLOPE * v; }

// float atomic max via monotone bit ordering (init must be -inf or a real value)
__device__ __forceinline__ void atomicMaxF(float* a, float v) {
    if (v >= 0.f) atomicMax((int*)a, __float_as_int(v));
    else          atomicMin((unsigned int*)a, __float_as_uint(v));
}

// ---------------------------------------------------------------- TDM helper
// 2-D tensor_load_to_lds: D# built per CDNA5 ISA ch.8 (groups 0/1; later
// groups zero -> dims beyond 2D unused). data_size=4B, dims/strides in
// elements. This toolchain exposes the 6-arg builtin form:
//   (uint32x4 g0, int32x8 g1, int32x4, int32x4, int32x8, i32 cpol)
__device__ __forceinline__ void tdm_load_2d(unsigned lds_addr, const void* gptr,
                                            unsigned dim0, unsigned dim1,
                                            unsigned tile0, unsigned tile1,
                                            unsigned stride0) {
    unsigned long long ga = (unsigned long long)(uintptr_t)gptr;
    v4u g0;
    g0.x = 1u;                                               // count=1, user desc
    g0.y = lds_addr;                                         // LDS byte offset
    g0.z = (unsigned)(ga & 0xffffffffu);                     // global_addr[31:0]
    g0.w = (unsigned)((ga >> 32) & 0x01ffffffu) | (2u << 30); // addr[56:32] | type=2
    v8i g1;
    g1[0] = (int)(2u << 16);                                 // data_size = 4 bytes
    g1[1] = (int)((dim0 & 0xffffu) << 16);                   // tensor_dim0[15:0]
    g1[2] = (int)((dim0 >> 16) | ((dim1 & 0xffffu) << 16));  // dim0[31:16]|dim1[15:0]
    g1[3] = (int)((dim1 >> 16) | (tile0 << 16));             // dim1[31:16]|tile_dim0
    g1[4] = (int)(tile1 & 0xffffu);                          // tile_dim1 (tile_dim2=0)
    g1[5] = (int)stride0;                                    // tensor_dim0_stride[31:0]
    g1[6] = 0;
    g1[7] = 0;
    v4i z4 = {0, 0, 0, 0};
    v8i z8 = {0, 0, 0, 0, 0, 0, 0, 0};
    __builtin_amdgcn_tensor_load_to_lds(g0, g1, z4, z4, z8, 0);
}

// ---------------------------------------------------------------- utilities
__global__ void fill_kernel(float* __restrict__ p, float v, int n) {
    int i = blockIdx.x * blockDim.x + threadIdx.x;
    if (i < n) p[i] = v;
}

__global__ void copy2_kernel(const float* __restrict__ s, float* __restrict__ d0,
                             float* __restrict__ d1, int n) {
    int i = blockIdx.x * blockDim.x + threadIdx.x;
    if (i < n) { float v = s[i]; d0[i] = v; d1[i] = v; }
}

// ---------------------------------------------------------------- WMMA GEMM
// H[N,64] = X[N,64] @ W[64,64], f32, V_WMMA_F32_16X16X4_F32.
// Block = 128 threads = 4 waves; wave w owns column tile w (16 cols);
// block b owns row tile b (16 rows). N_NODES % 16 == 0 (100000 = 6250*16).
//
// Wave 0 stages W (16KB) and the contiguous 16x64 X tile (4KB) into LDS via
// the Tensor Data Mover, waits on TENSORcnt, then the block barriers and all
// waves build fragments from LDS (bank-conflict-free, no strided global A
// loads). f32 fragment layouts per ISA 7.12.2:
//  A 16x4 : lanes 0-15 -> M=lane, {K=k0,k0+1}; lanes 16-31 -> {k0+2,k0+3}
//  B 4x16 : lanes 0-15 -> N=lane, rows {k0,k0+1}; lanes 16-31 -> {k0+2,k0+3}
//  C 16x16: lane<16 -> N=lane, VGPR v = row v; lane>=16 -> row v+8
__global__ void gemm64_wmma_kernel(const float* __restrict__ X,
                                   const float* __restrict__ W,
                                   float* __restrict__ H) {
    __shared__ float lds_w[64 * 64];
    __shared__ float lds_x[16 * 64];

    const int lane = threadIdx.x & 31;
    const int wv   = threadIdx.x >> 5;
    const int row0 = blockIdx.x * 16;
    const int colB = wv * 16;
    const int half = lane >> 4;      // 0 or 1
    const int l    = lane & 15;

    if (threadIdx.x < 32) {          // wave 0 issues both TDM DMAs
        tdm_load_2d((unsigned)(uintptr_t)&lds_w[0], W,
                    64, 64, 64, 64, 64);
        tdm_load_2d((unsigned)(uintptr_t)&lds_x[0], X + (size_t)row0 * HIDC,
                    64, 16, 64, 16, 64);
        __builtin_amdgcn_s_wait_tensorcnt(0);
    }
    __syncthreads();

    v8f acc = {};
    const float* xr = lds_x + l * HIDC;
#pragma unroll
    for (int k0 = 0; k0 < HIDC; k0 += 4) {
        const int ka = k0 + (half << 1);
        v2f a = *(const v2f*)(xr + ka);            // A[M=row0+l][ka, ka+1]
        v2f b;
        b.x = lds_w[ka * HIDC + colB + l];         // B[ka  ][colB+l]
        b.y = lds_w[(ka + 1) * HIDC + colB + l];
        acc = __builtin_amdgcn_wmma_f32_16x16x4_f32(
                  false, a, false, b, (short)0, acc, false, false);
    }
    float* hrow = H + (size_t)(row0 + half * 8) * HIDC + colB + l;
#pragma unroll
    for (int v = 0; v < 8; ++v)
        hrow[(size_t)v * HIDC] = acc[v];
}

// ---------------------------------------------------------------- random walk
// out[row[e]] += x[col[e]]  (thread = (edge, group of 4 features))
__global__ void rw_scatter_kernel(const float* __restrict__ xin,
                                  float* __restrict__ xout,
                                  const long long* __restrict__ ei) {
    int t = blockIdx.x * blockDim.x + threadIdx.x;
    if (t >= N_EDGES * 16) return;
    int e = t >> 4;
    int f = (t & 15) << 2;
    long long r = ei[e];             // row  (scatter target)
    long long c = ei[N_EDGES + e];   // col  (gather source)
    const float4 s = *(const float4*)(xin + (size_t)c * HIDC + f);
    float* d = xout + (size_t)r * HIDC + f;
    atomicAdd(d + 0, s.x); atomicAdd(d + 1, s.y);
    atomicAdd(d + 2, s.z); atomicAdd(d + 3, s.w);
}

// ---------------------------------------------------------------- GAT
// pack gat_W[8,64,8] -> Wp[64,64] row-major (row=f, col=head*8+d)
__global__ void gat_pack_kernel(const float* __restrict__ gw, float* __restrict__ Wp) {
    int i = blockIdx.x * blockDim.x + threadIdx.x;
    if (i >= 64 * 64) return;
    int f = i >> 6, c = i & 63;
    int h = c >> 3, d = c & 7;
    Wp[i] = gw[h * (64 * 8) + f * 8 + d];
}

__global__ void gat_al_kernel(const float* __restrict__ hf,
                              const float* __restrict__ a_src,
                              const float* __restrict__ a_dst,
                              float* __restrict__ als, float* __restrict__ ald) {
    int t = blockIdx.x * blockDim.x + threadIdx.x;   // n*8+h
    if (t >= N_NODES * HEADS) return;
    int n = t >> 3, h = t & 7;
    const float* hp = hf + (size_t)n * HIDC + h * 8;
    float s = 0.f, d = 0.f;
#pragma unroll
    for (int k = 0; k < 8; ++k) { s += hp[k] * a_src[h * 8 + k]; d += hp[k] * a_dst[h * 8 + k]; }
    als[t] = s; ald[t] = d;
}

// init segment-max with the self-loop value
__global__ void gat_minit_kernel(const float* __restrict__ als,
                                 const float* __restrict__ ald,
                                 float* __restrict__ m) {
    int t = blockIdx.x * blockDim.x + threadIdx.x;
    if (t >= N_NODES * HEADS) return;
    m[t] = leaky(als[t] + ald[t]);
}

__global__ void gat_max_kernel(const float* __restrict__ als,
                               const float* __restrict__ ald,
                               const long long* __restrict__ ei,
                               float* __restrict__ m) {
    int t = blockIdx.x * blockDim.x + threadIdx.x;   // e*8+h
    if (t >= N_EDGES * HEADS) return;
    int e = t >> 3, h = t & 7;
    long long s = ei[e], d = ei[N_EDGES + e];        // src=row, dst=col
    float ev = leaky(als[s * 8 + h] + ald[d * 8 + h]);
    atomicMaxF(&m[d * 8 + h], ev);
}

__global__ void gat_dinit_kernel(const float* __restrict__ als,
                                 const float* __restrict__ ald,
                                 const float* __restrict__ m,
                                 float* __restrict__ den) {
    int t = blockIdx.x * blockDim.x + threadIdx.x;
    if (t >= N_NODES * HEADS) return;
    den[t] = __expf(leaky(als[t] + ald[t]) - m[t]);  // self-loop term
}

__global__ void gat_dsum_kernel(const float* __restrict__ als,
                                const float* __restrict__ ald,
                                const long long* __restrict__ ei,
                                const float* __restrict__ m,
                                float* __restrict__ den) {
    int t = blockIdx.x * blockDim.x + threadIdx.x;
    if (t >= N_EDGES * HEADS) return;
    int e = t >> 3, h = t & 7;
    long long s = ei[e], d = ei[N_EDGES + e];
    float ev = leaky(als[s * 8 + h] + ald[d * 8 + h]);
    atomicAdd(&den[d * 8 + h], __expf(ev - m[d * 8 + h]));
}

// out init = bias + self-loop alpha * h[n]
__global__ void gat_oinit_kernel(const float* __restrict__ hf,
                                 const float* __restrict__ als,
                                 const float* __restrict__ ald,
                                 const float* __restrict__ m,
                                 const float* __restrict__ den,
                                 const float* __restrict__ gb,
                                 float* __restrict__ out) {
    int t = blockIdx.x * blockDim.x + threadIdx.x;   // n*64+c
    if (t >= N_NODES * HIDC) return;
    int n = t >> 6, c = t & 63, h = c >> 3;
    float ev = leaky(als[n * 8 + h] + ald[n * 8 + h]);
    float alpha = __expf(ev - m[n * 8 + h]) / den[n * 8 + h];
    out[t] = gb[c] + alpha * hf[t];
}

__global__ void gat_osum_kernel(const float* __restrict__ hf,
                                const float* __restrict__ als,
                                const float* __restrict__ ald,
                                const long long* __restrict__ ei,
                                const float* __restrict__ m,
                                const float* __restrict__ den,
                                float* __restrict__ out) {
    int t = blockIdx.x * blockDim.x + threadIdx.x;   // e*8+h
    if (t >= N_EDGES * HEADS) return;
    int e = t >> 3, h = t & 7;
    long long s = ei[e], d = ei[N_EDGES + e];
    float ev = leaky(als[s * 8 + h] + ald[d * 8 + h]);
    float alpha = __expf(ev - m[d * 8 + h]) / den[d * 8 + h];
    const float* hp = hf + (size_t)s * HIDC + h * 8;
    float* op = out + (size_t)d * HIDC + h * 8;
#pragma unroll
    for (int k = 0; k < 8; ++k) atomicAdd(op + k, alpha * hp[k]);
}

// ---------------------------------------------------------------- GCN
__global__ void deg_edge_kernel(const long long* __restrict__ ei, float* __restrict__ deg) {
    int e = blockIdx.x * blockDim.x + threadIdx.x;
    if (e >= N_EDGES) return;
    atomicAdd(&deg[ei[N_EDGES + e]], 1.f);           // dst = col
}

__global__ void dis_kernel(float* __restrict__ deg) {
    int n = blockIdx.x * blockDim.x + threadIdx.x;
    if (n < N_NODES) deg[n] = rsqrtf(deg[n]);        // deg >= 1 (self loop)
}

__global__ void gcn_self_kernel(const float* __restrict__ h, const float* __restrict__ dis,
                                float* __restrict__ out) {
    int t = blockIdx.x * blockDim.x + threadIdx.x;
    if (t >= N_NODES * HIDC) return;
    int n = t >> 6;
    out[t] = dis[n] * dis[n] * h[t];
}

__global__ void gcn_edge_kernel(const float* __restrict__ h, const float* __restrict__ dis,
                                const long long* __restrict__ ei, float* __restrict__ out) {
    int t = blockIdx.x * blockDim.x + threadIdx.x;
    if (t >= N_EDGES * 16) return;
    int e = t >> 4;
    int f = (t & 15) << 2;
    long long s = ei[e], d = ei[N_EDGES + e];
    float w = dis[s] * dis[d];
    const float4 hv = *(const float4*)(h + (size_t)s * HIDC + f);
    float* op = out + (size_t)d * HIDC + f;
    atomicAdd(op + 0, w * hv.x); atomicAdd(op + 1, w * hv.y);
    atomicAdd(op + 2, w * hv.z); atomicAdd(op + 3, w * hv.w);
}

// x = relu(out + b [+ res]); res updated to pre-relu value when use_res
__global__ void gcn_ep1_kernel(const float* __restrict__ out, const float* __restrict__ b,
                               float* __restrict__ res, float* __restrict__ x, int use_res) {
    int t = blockIdx.x * blockDim.x + threadIdx.x;
    if (t >= N_NODES * HIDC) return;
    float v = out[t] + b[t & 63];
    if (use_res) { v += res[t]; res[t] = v; }
    x[t] = fmaxf(v, 0.f);
}

// x += out + b
__global__ void gcn_ep2_kernel(const float* __restrict__ out, const float* __restrict__ b,
                               float* __restrict__ x) {
    int t = blockIdx.x * blockDim.x + threadIdx.x;
    if (t >= N_NODES * HIDC) return;
    x[t] += out[t] + b[t & 63];
}

// ---------------------------------------------------------------- final linear
__global__ void fc_kernel(const float* __restrict__ x, const float* __restrict__ W,
                          const float* __restrict__ b, float* __restrict__ out) {
    int n = blockIdx.x * blockDim.x + threadIdx.x;
    if (n >= N_NODES) return;
    float acc[OUTC];
#pragma unroll
    for (int o = 0; o < OUTC; ++o) acc[o] = b[o];
    const float* xr = x + (size_t)n * HIDC;
#pragma unroll 8
    for (int f = 0; f < HIDC; ++f) {
        float xv = xr[f];
#pragma unroll
        for (int o = 0; o < OUTC; ++o) acc[o] += xv * W[f * OUTC + o];
    }
    float* orow = out + (size_t)n * OUTC;
#pragma unroll
    for (int o = 0; o < OUTC; ++o) orow[o] = acc[o];
}

// ================================================================ launch
extern "C" void kernel_launch(void* const* d_in, const int* in_sizes, int n_in,
                              void* d_out, int out_size, void* d_ws, size_t ws_size,
                              hipStream_t stream) {
    (void)in_sizes; (void)n_in; (void)out_size; (void)ws_size;
    const float*     x_in   = (const float*)d_in[0];
    const long long* ei     = (const long long*)d_in[1];   // int64 [2,E]
    const float*     gat_W  = (const float*)d_in[2];
    const float*     gat_as = (const float*)d_in[3];
    const float*     gat_ad = (const float*)d_in[4];
    const float*     gat_b  = (const float*)d_in[5];
    const float*     gcn_W  = (const float*)d_in[6];
    const float*     gcn_b  = (const float*)d_in[7];
    const float*     fc_W   = (const float*)d_in[8];
    const float*     fc_b   = (const float*)d_in[9];

    const size_t NF = (size_t)N_NODES * HIDC;
    const size_t NH = (size_t)N_NODES * HEADS;
    float* ws   = (float*)d_ws;
    float* bufA = ws; ws += NF;      // current x
    float* bufH = ws; ws += NF;      // gemm output / rw ping-pong
    float* bufO = ws; ws += NF;      // aggregation output
    float* bufR = ws; ws += NF;      // initial_residual
    float* als  = ws; ws += NH;
    float* ald  = ws; ws += NH;
    float* mbuf = ws; ws += NH;
    float* dbuf = ws; ws += NH;
    float* dis  = ws; ws += N_NODES;
    float* Wp   = ws; ws += 64 * 64;

    const int TB = 256;
    auto g = [](long long n) { return (int)((n + 255) / 256); };

    // ---- 1) random walk: 5 ping-pong scatter steps (x_in -> ... -> bufA)
    const float* src = x_in;
    float* dst = bufA;
    for (int s = 0; s < RW_STEPS; ++s) {
        fill_kernel<<<g((long long)NF), TB, 0, stream>>>(dst, 0.f, (int)NF);
        rw_scatter_kernel<<<g((long long)N_EDGES * 16), TB, 0, stream>>>(src, dst, ei);
        src = dst;
        dst = (dst == bufA) ? bufH : bufA;
    }
    // RW_STEPS is odd -> result in bufA

    // ---- 2) GAT (8 heads)
    gat_pack_kernel<<<g(64 * 64), TB, 0, stream>>>(gat_W, Wp);
    gemm64_wmma_kernel<<<N_NODES / 16, 128, 0, stream>>>(bufA, Wp, bufH);   // h = x @ Wp
    gat_al_kernel<<<g((long long)NH), TB, 0, stream>>>(bufH, gat_as, gat_ad, als, ald);
    gat_minit_kernel<<<g((long long)NH), TB, 0, stream>>>(als, ald, mbuf);
    gat_max_kernel<<<g((long long)N_EDGES * HEADS), TB, 0, stream>>>(als, ald, ei, mbuf);
    gat_dinit_kernel<<<g((long long)NH), TB, 0, stream>>>(als, ald, mbuf, dbuf);
    gat_dsum_kernel<<<g((long long)N_EDGES * HEADS), TB, 0, stream>>>(als, ald, ei, mbuf, dbuf);
    gat_oinit_kernel<<<g((long long)NF), TB, 0, stream>>>(bufH, als, ald, mbuf, dbuf, gat_b, bufO);
    gat_osum_kernel<<<g((long long)N_EDGES * HEADS), TB, 0, stream>>>(bufH, als, ald, ei, mbuf, dbuf, bufO);
    // x = GAT out; initial_residual = x
    copy2_kernel<<<g((long long)NF), TB, 0, stream>>>(bufO, bufA, bufR, (int)NF);

    // ---- 3) GCN normalization (layer-invariant): dis = rsqrt(1 + in-degree)
    fill_kernel<<<g((long long)N_NODES), TB, 0, stream>>>(dis, 1.f, N_NODES);
    deg_edge_kernel<<<g((long long)N_EDGES), TB, 0, stream>>>(ei, dis);
    dis_kernel<<<g((long long)N_NODES), TB, 0, stream>>>(dis);

    // ---- 12 residual GCN layers (with the reference's W[i-1] quirk)
    for (int i = 0; i < N_LAYERS; ++i) {
        gemm64_wmma_kernel<<<N_NODES / 16, 128, 0, stream>>>(bufA, gcn_W + (size_t)i * 4096, bufH);
        gcn_self_kernel<<<g((long long)NF), TB, 0, stream>>>(bufH, dis, bufO);
        gcn_edge_kernel<<<g((long long)N_EDGES * 16), TB, 0, stream>>>(bufH, dis, ei, bufO);
        gcn_ep1_kernel<<<g((long long)NF), TB, 0, stream>>>(bufO, gcn_b + (size_t)i * HIDC,
                                                            bufR, bufA, i > 0 ? 1 : 0);
        if (i < N_LAYERS - 1) {
            int j = (i + N_LAYERS - 1) % N_LAYERS;   // i-1, wraps to 11 at i==0
            gemm64_wmma_kernel<<<N_NODES / 16, 128, 0, stream>>>(bufA, gcn_W + (size_t)j * 4096, bufH);
            gcn_self_kernel<<<g((long long)NF), TB, 0, stream>>>(bufH, dis, bufO);
            gcn_edge_kernel<<<g((long long)N_EDGES * 16), TB, 0, stream>>>(bufH, dis, ei, bufO);
            gcn_ep2_kernel<<<g((long long)NF), TB, 0, stream>>>(bufO, gcn_b + (size_t)j * HIDC, bufA);
        }
    }

    // ---- 4) final linear
    fc_kernel<<<g((long long)N_NODES), TB, 0, stream>>>(bufA, fc_W, fc_b, (float*)d_out);
}